// ModelGraphIsomorphismOne_51049981280515
// MI455X (gfx1250) — compile-verified
//
#include <hip/hip_runtime.h>

typedef float v2f __attribute__((ext_vector_type(2)));
typedef float v8f __attribute__((ext_vector_type(8)));

#define N_NODES 20000
#define N_EDGES 320000
#define BERT_H  768
#define GIN_H   256

// ---------------------------------------------------------------------------
// rst = h (h may be a column-slice of the cat buffer; ldh = row stride in floats)
// ---------------------------------------------------------------------------
__global__ void init_rst_kernel(const float* __restrict__ h, int ldh,
                                float* __restrict__ rst, int D, int chunks) {
  int idx = blockIdx.x * blockDim.x + threadIdx.x;
  int total = N_NODES * chunks;           // chunks = D/4
  if (idx >= total) return;
  int row = idx / chunks;
  int c   = idx - row * chunks;
  const float4* sp = (const float4*)(h + (size_t)row * ldh);
  float4 v = sp[c];
  float4* dp = (float4*)(rst + (size_t)row * D);
  dp[c] = v;
}

// ---------------------------------------------------------------------------
// rst[dst[e]] += w[e] * h[src[e]]   (one wave32 per edge, float4 chunks)
// ---------------------------------------------------------------------------
__global__ void scatter_kernel(const float* __restrict__ h, int ldh,
                               const float* __restrict__ ew,
                               const int* __restrict__ src,
                               const int* __restrict__ dst,
                               float* __restrict__ rst, int D) {
  int wid  = (blockIdx.x * blockDim.x + threadIdx.x) >> 5;  // edge id
  int lane = threadIdx.x & 31;
  if (wid >= N_EDGES) return;
  int   s = src[wid];
  int   d = dst[wid];
  float w = ew[wid];
  const float4* hp = (const float4*)(h + (size_t)s * ldh);
  float* rp = rst + (size_t)d * D;
  int nd4 = D >> 2;
  for (int c = lane; c < nd4; c += 32) {
    float4 v = hp[c];
    atomicAdd(rp + 4 * c + 0, v.x * w);
    atomicAdd(rp + 4 * c + 1, v.y * w);
    atomicAdd(rp + 4 * c + 2, v.z * w);
    atomicAdd(rp + 4 * c + 3, v.w * w);
  }
}

// ---------------------------------------------------------------------------
// out[M,256] = (relu?) (A[M,K] @ W[K,256] + bias)
// Wave computes a 32(M) x 64(N) tile with V_WMMA_F32_16X16X4_F32: two M-tiles
// share every B (weight) fetch, halving the load:wmma ratio vs 16x64.
// A layout per ISA: lanes 0-15 hold rows M=lane with K=k,k+1 in v0,v1;
// lanes 16-31 same rows with K=k+2,k+3. B mirrored (row=K striped by lane).
// C/D: VGPR r holds M=r (lanes 0-15) and M=8+r (lanes 16-31), N=lane&15.
// Block = 4 waves (128 thr): all waves share the same 32 A-rows, each wave
// owns a distinct 64-column group => block covers the full N=256.
// ---------------------------------------------------------------------------
__global__ void __launch_bounds__(128)
gin_gemm_relu_kernel(const float* __restrict__ A, int lda,
                     const float* __restrict__ W,
                     const float* __restrict__ bias,
                     float* __restrict__ out, int ldo,
                     int K, int do_relu) {
  const int tid  = threadIdx.x;
  const int wave = tid >> 5;          // 0..3  -> column group
  const int lane = tid & 31;
  const int lo   = lane & 15;
  const int hi   = lane >> 4;

  const int mbase = blockIdx.x * 32;  // 0..19968, step 32
  const int nbase = wave * 64;        // 0,64,128,192

  v8f acc0[4], acc1[4];
#pragma unroll
  for (int j = 0; j < 4; ++j) {
    float bv = bias[nbase + j * 16 + lo];
#pragma unroll
    for (int r = 0; r < 8; ++r) { acc0[j][r] = bv; acc1[j][r] = bv; }
  }

  const float* Arow0 = A + (size_t)(mbase + lo) * lda;
  const float* Arow1 = A + (size_t)(mbase + 16 + lo) * lda;

#pragma unroll 2
  for (int k = 0; k < K; k += 4) {
    const int kk = k + 2 * hi;
    v2f a0 = *(const v2f*)(Arow0 + kk);
    v2f a1 = *(const v2f*)(Arow1 + kk);
#pragma unroll
    for (int j = 0; j < 4; ++j) {
      const int col = nbase + j * 16 + lo;
      v2f b;
      b.x = W[(size_t)kk * GIN_H + col];
      b.y = W[(size_t)(kk + 1) * GIN_H + col];
      acc0[j] = __builtin_amdgcn_wmma_f32_16x16x4_f32(
          false, a0, false, b, (short)0, acc0[j], false, false);
      acc1[j] = __builtin_amdgcn_wmma_f32_16x16x4_f32(
          false, a1, false, b, (short)0, acc1[j], false, false);
    }
  }

#pragma unroll
  for (int j = 0; j < 4; ++j) {
    const int col = nbase + j * 16 + lo;
#pragma unroll
    for (int r = 0; r < 8; ++r) {
      float v0 = acc0[j][r];
      float v1 = acc1[j][r];
      if (do_relu) { v0 = fmaxf(v0, 0.0f); v1 = fmaxf(v1, 0.0f); }
      out[(size_t)(mbase + r + 8 * hi) * ldo + col]      = v0;
      out[(size_t)(mbase + 16 + r + 8 * hi) * ldo + col] = v1;
    }
  }
}

// ---------------------------------------------------------------------------
// Orchestration.  Workspace: rst [20000*768] f32  |  cat [20000*768] f32
// (122.9 MB total).  Layer l output lands in cat columns [l*256,(l+1)*256).
// ---------------------------------------------------------------------------
extern "C" void kernel_launch(void* const* d_in, const int* in_sizes, int n_in,
                              void* d_out, int out_size, void* d_ws, size_t ws_size,
                              hipStream_t stream) {
  const float* node_embed = (const float*)d_in[0];
  const float* ew = (const float*)d_in[1];
  const float* W0 = (const float*)d_in[2];
  const float* b0 = (const float*)d_in[3];
  const float* W1 = (const float*)d_in[4];
  const float* b1 = (const float*)d_in[5];
  const float* W2 = (const float*)d_in[6];
  const float* b2 = (const float*)d_in[7];
  const float* Wr = (const float*)d_in[8];
  const float* br = (const float*)d_in[9];
  const int* src = (const int*)d_in[10];
  const int* dst = (const int*)d_in[11];

  float* rst = (float*)d_ws;
  float* cat = (float*)((char*)d_ws + (size_t)N_NODES * BERT_H * sizeof(float));

  const int GEMM_BLOCKS = N_NODES / 32;  // 625 blocks x 4 waves (32x256 per block)
  const int SCAT_BLOCKS = N_EDGES / 8;   // 1 wave per edge, 8 waves/block

  // ---- layer 0 : D = 768 -------------------------------------------------
  {
    const int D = BERT_H, chunks = D / 4, total = N_NODES * chunks;
    init_rst_kernel<<<(total + 255) / 256, 256, 0, stream>>>(node_embed, BERT_H, rst, D, chunks);
    scatter_kernel<<<SCAT_BLOCKS, 256, 0, stream>>>(node_embed, BERT_H, ew, src, dst, rst, D);
    gin_gemm_relu_kernel<<<GEMM_BLOCKS, 128, 0, stream>>>(rst, D, W0, b0, cat + 0, BERT_H, D, 1);
  }
  // ---- layer 1 : D = 256 -------------------------------------------------
  {
    const int D = GIN_H, chunks = D / 4, total = N_NODES * chunks;
    init_rst_kernel<<<(total + 255) / 256, 256, 0, stream>>>(cat + 0, BERT_H, rst, D, chunks);
    scatter_kernel<<<SCAT_BLOCKS, 256, 0, stream>>>(cat + 0, BERT_H, ew, src, dst, rst, D);
    gin_gemm_relu_kernel<<<GEMM_BLOCKS, 128, 0, stream>>>(rst, D, W1, b1, cat + GIN_H, BERT_H, D, 1);
  }
  // ---- layer 2 : D = 256 -------------------------------------------------
  {
    const int D = GIN_H, chunks = D / 4, total = N_NODES * chunks;
    init_rst_kernel<<<(total + 255) / 256, 256, 0, stream>>>(cat + GIN_H, BERT_H, rst, D, chunks);
    scatter_kernel<<<SCAT_BLOCKS, 256, 0, stream>>>(cat + GIN_H, BERT_H, ew, src, dst, rst, D);
    gin_gemm_relu_kernel<<<GEMM_BLOCKS, 128, 0, stream>>>(rst, D, W2, b2, cat + 2 * GIN_H, BERT_H, D, 1);
  }
  // ---- readout: out = relu(cat) @ Wr + br ; cat >= 0 already ------------
  gin_gemm_relu_kernel<<<GEMM_BLOCKS, 128, 0, stream>>>(cat, BERT_H, Wr, br,
                                                        (float*)d_out, GIN_H, BERT_H, 0);
}